// ANIFeat_29746943492212
// MI455X (gfx1250) — compile-verified
//
#include <hip/hip_runtime.h>
#include <hip/hip_bf16.h>

#define BOHR  0.52917721092f
#define RC_   4.6f
#define AC_   3.1f
#define RLn   32
#define ALn   8
#define NBAT  4
#define NAT   23
#define NCASE 5
#define NPAIR 15
#define NPP   (NAT*NAT)     /* 529 */
#define NPPAD 532           /* padded to multiple of 4 */
#define NKC   133           /* ceil(529/4) K-chunks of 4 */
#define FEAT  (1 + NCASE*RLn + NPAIR*ALn*ALn)   /* 1121 */
#define PI_F  3.14159265358979323846f

typedef float v2f __attribute__((ext_vector_type(2)));
typedef float v8f __attribute__((ext_vector_type(8)));

__device__ __constant__ int c_pj[16]    = {0,0,0,0,0,1,1,1,1,2,2,2,3,3,4,0};
__device__ __constant__ int c_pk[16]    = {0,1,2,3,4,1,2,3,4,2,3,4,3,4,4,0};
__device__ __constant__ int c_cases[5]  = {1,6,7,8,16};

__global__ __launch_bounds__(128)
void ani_feat_kernel(const float* __restrict__ coords,
                     const int*   __restrict__ z,
                     float*       __restrict__ out)
{
    const int bi  = blockIdx.x;       // 0 .. B*N-1 (92)
    const int b   = bi / NAT;
    const int i   = bi % NAT;
    const int tid = threadIdx.x;

    __shared__ float s_cx[NAT], s_cy[NAT], s_cz[NAT];
    __shared__ float s_rx[NAT], s_ry[NAT], s_rz[NAT];
    __shared__ float s_d[NAT],  s_fac[NAT], s_frc[NAT];
    __shared__ float s_sel[24][NCASE];                  // row 23 = zero pad
    __shared__ float s_px[NPPAD], s_ps[NPPAD], s_pr[NPPAD], s_pf[NPPAD];

    // ---- load coords (pre-scaled by BOHR) + selection one-hot ----
    if (tid < NAT) {
        s_cx[tid] = coords[(b*NAT + tid)*3 + 0] * BOHR;
        s_cy[tid] = coords[(b*NAT + tid)*3 + 1] * BOHR;
        s_cz[tid] = coords[(b*NAT + tid)*3 + 2] * BOHR;
    }
    if (tid < 24*NCASE) {
        int j = tid / NCASE, t = tid % NCASE;
        float v = 0.0f;
        if (j < NAT) v = (z[b*NAT + j] == c_cases[t]) ? 1.0f : 0.0f;
        s_sel[j][t] = v;
    }
    __syncthreads();

    // ---- per-j: r_ij = c_j - c_i, distance, cutoffs ----
    if (tid < NAT) {
        const int j = tid;
        float dx = s_cx[j] - s_cx[i];
        float dy = s_cy[j] - s_cy[i];
        float dz = s_cz[j] - s_cz[i];
        float d  = sqrtf(dx*dx + dy*dy + dz*dz + 1e-7f);
        s_rx[j] = dx; s_ry[j] = dy; s_rz[j] = dz; s_d[j] = d;
        float fac = 0.0f, frc = 0.0f;
        if (j != i) {
            if (d < AC_) fac = 0.5f * (__cosf(PI_F * d / AC_) + 1.0f);
            if (d < RC_) frc = 0.5f * (__cosf(PI_F * d / RC_) + 1.0f);
        }
        s_fac[j] = fac; s_frc[j] = frc;
    }
    __syncthreads();

    // ---- per (j,k) pair scalars: cos(theta), sin(theta), (Rij+Rik)/2, 2*f*f ----
    for (int jk = tid; jk < NPPAD; jk += 128) {
        float x = 0.0f, sn = 0.0f, rv = 0.0f, fp = 0.0f;
        if (jk < NPP) {
            int j = jk / NAT, k = jk - j*NAT;
            float vm    = s_rx[j]*s_rx[k] + s_ry[j]*s_ry[k] + s_rz[j]*s_rz[k];
            float denom = s_d[j]*s_d[k] + 1e-5f;
            float sflag = (s_fac[j] > 0.0f && s_fac[k] > 0.0f) ? 1.0f : 0.0f;
            x  = vm * sflag / denom;                 // cos(theta), |x| < 1
            sn = sqrtf(fmaxf(0.0f, 1.0f - x*x));     // sin(theta) >= 0
            rv = 0.5f * (s_d[j] + s_d[k]);
            fp = 2.0f * s_fac[j] * s_fac[k];
        }
        s_px[jk] = x; s_ps[jk] = sn; s_pr[jk] = rv; s_pf[jk] = fp;
    }
    __syncthreads();

    float* feat = out + (size_t)(b*NAT + i) * FEAT;
    if (tid == 0) feat[0] = (float)z[b*NAT + i];

    // ---- radial symmetry functions (VALU, 160 outputs) ----
    const float dRs   = RC_ / (RLn - 1);
    const float ITA_R = 3.0f / (dRs * dRs);
    for (int idx = tid; idx < NCASE*RLn; idx += 128) {
        int t = idx / RLn, r = idx % RLn;
        float rs = (float)r * dRs;
        float acc = 0.0f;
        for (int j = 0; j < NAT; ++j) {
            float dd = s_d[j] - rs;
            acc += __expf(-ITA_R * dd * dd) * s_frc[j] * s_sel[j][t];
        }
        feat[1 + idx] = acc;
    }

    // ---- angular part: D(16x16) = A^T(15x529) x B(529x64-tile) via WMMA f32 16x16x4 ----
    const int lane  = tid & 31;
    const int wave  = tid >> 5;          // 4 waves: N-tiles l in [16w, 16w+16)
    const int nn    = lane & 15;
    const int khalf = lane >> 4;         // K half: lanes 0-15 -> K {0,1}, 16-31 -> K {2,3}

    // A-side lane constants (M = pair class p = nn)
    const float maskp = (nn < NPAIR) ? 1.0f : 0.0f;
    const int pj = c_pj[nn];
    const int pk = c_pk[nn];

    // B-side lane constants (N = angular channel l = 16*wave + nn; l = rs*8 + th)
    const int   l     = wave*16 + nn;
    const int   thi   = l & 7;
    const int   rsi   = l >> 3;
    const float dRa   = AC_ / (ALn - 1);
    const float ITA_A = 3.0f / (dRa * dRa);
    const float rsA   = (float)rsi * dRa;
    float st, ct;
    __sincosf(PI_F * (float)thi / 7.0f, &st, &ct);

    v8f acc = {0.f,0.f,0.f,0.f,0.f,0.f,0.f,0.f};

    for (int kc = 0; kc < NKC; ++kc) {
        const int jk0 = kc*4 + khalf*2;
        v2f av, bv;
        #pragma unroll
        for (int kk = 0; kk < 2; ++kk) {
            const int jk = jk0 + kk;                       // < 532
            const int j  = jk / NAT;                       // <= 23 (pad row)
            const int k  = jk - j*NAT;
            // A: sel_pairs[jk, p]
            av[kk] = s_sel[j][pj] * s_sel[k][pk] * maskp;
            // B: out[jk, l] = ((1+cos(th-th_m))/2)^64 * exp(-ita*(ravg-Rs)^2) * 2*f*f
            float x  = s_px[jk], sn = s_ps[jk], rv = s_pr[jk], fp = s_pf[jk];
            float c1 = 0.5f * (1.0f + x*ct + sn*st);
            float c2 = c1*c1; c2*=c2; c2*=c2; c2*=c2; c2*=c2; c2*=c2;   // ^64
            float dr = rv - rsA;
            bv[kk] = c2 * __expf(-ITA_A * dr * dr) * fp;
        }
        acc = __builtin_amdgcn_wmma_f32_16x16x4_f32(
                  false, av, false, bv, (short)0, acc, false, false);
    }

    // D layout: lane holds col N = nn, rows M = v + 8*khalf
    #pragma unroll
    for (int v = 0; v < 8; ++v) {
        const int pp = v + 8*khalf;
        if (pp < NPAIR) {
            feat[1 + NCASE*RLn + pp*(ALn*ALn) + l] = acc[v];
        }
    }
}

extern "C" void kernel_launch(void* const* d_in, const int* in_sizes, int n_in,
                              void* d_out, int out_size, void* d_ws, size_t ws_size,
                              hipStream_t stream) {
    const float* coords = (const float*)d_in[0];   // (4,23,3) f32
    const int*   zatom  = (const int*)d_in[1];     // (4,23)   i32
    float*       outp   = (float*)d_out;           // (4,23,1121) f32
    (void)in_sizes; (void)n_in; (void)out_size; (void)d_ws; (void)ws_size;
    ani_feat_kernel<<<NBAT*NAT, 128, 0, stream>>>(coords, zatom, outp);
}